// BatchedPrecisionAttentionBlock_22204980920933
// MI455X (gfx1250) — compile-verified
//
#include <hip/hip_runtime.h>
#include <math.h>

#define Bb 2
#define Ss 512
#define Hh 64
#define Dd 64
#define NUc 0.5f
#define DELTAc 1.0f
#define ETAc 1.0f

typedef float v2f __attribute__((ext_vector_type(2)));
typedef float v8f __attribute__((ext_vector_type(8)));

static __device__ __forceinline__ v8f wmma4(v2f a, v2f b, v8f c) {
  // V_WMMA_F32_16X16X4_F32 : D(16x16 f32) = A(16x4 f32) x B(4x16 f32) + C
  return __builtin_amdgcn_wmma_f32_16x16x4_f32(false, a, false, b, (short)0, c, false, false);
}

// ---------------------------------------------------------------------------
// Complex projection: Y(B,2,S,Dout) = X(B,2,S,Hin) @ W(2,Dout,Hin)^T + bias
// One wave per 16x16 output tile. 256 waves total (B * S/16 * Dout/16).
// ---------------------------------------------------------------------------
__global__ void cproj_kernel(const float* __restrict__ X, const float* __restrict__ W,
                             const float* __restrict__ bias, float* __restrict__ Y) {
  const int lane = threadIdx.x & 31;
  const int wid  = (blockIdx.x * blockDim.x + threadIdx.x) >> 5;
  const int b       = wid >> 7;          // 2
  const int rowTile = (wid >> 2) & 31;   // 32
  const int colTile = wid & 3;           // 4
  const int half = lane >> 4;
  const int lr16 = lane & 15;

  const int m = rowTile * 16 + lr16;     // A-matrix row (M)
  const int n = colTile * 16 + lr16;     // B-matrix col (N)

  const float* Xr = X + ((size_t)(b * 2 + 0) * Ss + m) * Hh;
  const float* Xi = X + ((size_t)(b * 2 + 1) * Ss + m) * Hh;
  const float* Wr = W + (size_t)n * Hh;              // plane 0, row n
  const float* Wi = W + (size_t)(Dd + n) * Hh;       // plane 1 (stride Dd*Hh)

  v8f accR = {}; v8f accI = {};
  for (int k0 = 0; k0 < Hh; k0 += 4) {
    const int ka = k0 + half * 2;
    v2f ar = __builtin_nontemporal_load((const v2f*)(Xr + ka));  // X read once
    v2f ai = __builtin_nontemporal_load((const v2f*)(Xi + ka));
    v2f br = *(const v2f*)(Wr + ka);
    v2f bi = *(const v2f*)(Wi + ka);
    v2f nbi = -bi;                         // f32 WMMA has no A/B NEG modifier
    accR = wmma4(ar, br,  accR);           // Yr += Xr*Wr^T
    accR = wmma4(ai, nbi, accR);           // Yr -= Xi*Wi^T
    accI = wmma4(ar, bi,  accI);           // Yi += Xr*Wi^T
    accI = wmma4(ai, br,  accI);           // Yi += Xi*Wr^T
  }
  const float br_ = bias[n];
  const float bi_ = bias[Dd + n];
  float* Yr = Y + (size_t)(b * 2 + 0) * Ss * Dd;
  float* Yi = Y + (size_t)(b * 2 + 1) * Ss * Dd;
#pragma unroll
  for (int v = 0; v < 8; ++v) {
    const int row = rowTile * 16 + (half ? v + 8 : v);
    const int col = n;
    Yr[(size_t)row * Dd + col] = accR[v] + br_;
    Yi[(size_t)row * Dd + col] = accI[v] + bi_;
  }
}

// ---------------------------------------------------------------------------
// Heavy pairwise kernel: one block (8 waves) per (b, i) row.
// Wave w handles j = w, w+8, ...; lane l handles d = l and d = l+32
// (conjugate pair -> one exp + one sincos covers both d's).
// Single pass: streams Xhat with NT stores, buffers unnormalized A in 128KB
// dynamic LDS, accumulates sumA and sum(A*Xhat) so est_v needs no re-read.
// Qij/Xhat are written non-temporal: 400MB of streaming output must not
// evict the reused Q/K/V rows from the 192MB L2.
// ---------------------------------------------------------------------------
__global__ void attn_kernel(const float* __restrict__ Qw, const float* __restrict__ Kw,
                            const float* __restrict__ Vw, const float* __restrict__ tm,
                            const float* __restrict__ lambda1,
                            const float* __restrict__ lamOm, const float* __restrict__ lamOm0,
                            const float* __restrict__ lamGam, const float* __restrict__ lamC,
                            float* __restrict__ Qij, float* __restrict__ Xhat,
                            float* __restrict__ EV) {
  extern __shared__ float smem[];
  float* A_lds = smem;                     // 512*64
  float* psA   = smem + Ss * Dd;           // 8*64
  float* pWr   = psA + 8 * Dd;             // 8*64
  float* pWi   = pWr + 8 * Dd;             // 8*64
  float* invA  = pWi + 8 * Dd;             // 64

  const int b   = blockIdx.x >> 9;
  const int i   = blockIdx.x & 511;
  const int tid = threadIdx.x;
  const int w   = tid >> 5;
  const int l   = tid & 31;
  const int d1 = l, d2 = l + 32;

  // per-lane spectral constants (d2 is conjugate of d1)
  const float lamr = lambda1[l];
  const float lami = lambda1[32 + l];
  const float re  = -fabsf(lamr);
  float om1 = lamOm[d1];  om1 *= om1;   float om2 = lamOm[d2];  om2 *= om2;
  float o01 = lamOm0[d1]; o01 *= o01;   float o02 = lamOm0[d2]; o02 *= o02;
  float g1  = lamGam[d1]; g1  *= g1;    float g2  = lamGam[d2]; g2  *= g2;
  const float C1 = lamC[d1], C2 = lamC[d2];

  const float ti = tm[i];
  const size_t qb0 = ((size_t)(b * 2 + 0) * Ss + i) * Dd;
  const size_t qb1 = ((size_t)(b * 2 + 1) * Ss + i) * Dd;
  const float Qr1 = Qw[qb0 + d1], Qr2 = Qw[qb0 + d2];
  const float Qi1 = Qw[qb1 + d1], Qi2 = Qw[qb1 + d2];

  float sA1 = 0.f, sA2 = 0.f;
  float Wr1 = 0.f, Wr2 = 0.f, Wi1 = 0.f, Wi2 = 0.f;

  const size_t qijRow = ((size_t)b * Ss + i) * Ss;
  const size_t xbase0 = ((size_t)(b * 2 + 0) * Ss + i) * Ss * Dd;
  const size_t xbase1 = ((size_t)(b * 2 + 1) * Ss + i) * Ss * Dd;

  const v2f zz = {0.f, 0.f};
  for (int j = w; j < Ss; j += 8) {
    const size_t xoff0 = xbase0 + (size_t)j * Dd;
    const size_t xoff1 = xbase1 + (size_t)j * Dd;
    if (j > i) {                           // masked region: explicit NT zeros (b64)
      __builtin_nontemporal_store(zz, (v2f*)(Qij + (qijRow + j) * Dd + 2 * l));
      __builtin_nontemporal_store(zz, (v2f*)(Xhat + xoff0 + 2 * l));
      __builtin_nontemporal_store(zz, (v2f*)(Xhat + xoff1 + 2 * l));
      continue;
    }
    const float dt = ti - tm[j];           // >= 0 inside mask
    const float dec = __expf(re * dt);
    float s1, c1; __sincosf(lami * dt, &s1, &c1);
    const float Kr1 = dec * c1, Ki1 = dec * s1;
    const float Kr2 = Kr1,      Ki2 = -Ki1;   // conjugate lane
    const float e2 = dec * dec;               // exp(2*re*|dt|)
    const float f  = (e2 - 1.0f) / (2.0f * re);
    const float P1 = 1.0f / (C1 * C1 * (e2 * o01 + om1 * f) + g1);
    const float P2 = 1.0f / (C2 * C2 * (e2 * o02 + om2 * f) + g2);

    const size_t vb0 = ((size_t)(b * 2 + 0) * Ss + j) * Dd;
    const size_t vb1 = ((size_t)(b * 2 + 1) * Ss + j) * Dd;
    const float Vr1  = Vw[vb0 + d1], Vr2  = Vw[vb0 + d2];
    const float Vi1  = Vw[vb1 + d1], Vi2  = Vw[vb1 + d2];
    const float Kkr1 = Kw[vb0 + d1], Kkr2 = Kw[vb0 + d2];
    const float Kki1 = Kw[vb1 + d1], Kki2 = Kw[vb1 + d2];

    const float Xr1  = Kr1 * Vr1 - Ki1 * Vi1,  Xi1v = Kr1 * Vi1 + Ki1 * Vr1;
    const float Xr2  = Kr2 * Vr2 - Ki2 * Vi2,  Xi2v = Kr2 * Vi2 + Ki2 * Vr2;
    const float Khr1 = Kr1 * Kkr1 - Ki1 * Kki1, Khi1 = Kr1 * Kki1 + Ki1 * Kkr1;
    const float Khr2 = Kr2 * Kkr2 - Ki2 * Kki2, Khi2 = Kr2 * Kki2 + Ki2 * Kkr2;
    const float Rr1 = Qr1 - C1 * Khr1, Ri1 = Qi1 - C1 * Khi1;
    const float Rr2 = Qr2 - C2 * Khr2, Ri2 = Qi2 - C2 * Khi2;

    float maha = P1 * (Rr1 * Rr1 + Ri1 * Ri1) + P2 * (Rr2 * Rr2 + Ri2 * Ri2);
#pragma unroll
    for (int off = 16; off > 0; off >>= 1) maha += __shfl_xor(maha, off, 32);
    const float inv_den = 1.0f / (1.0f + NUc * maha);
    const float A1 = P1 * inv_den, A2 = P2 * inv_den;

    __builtin_nontemporal_store(Xr1,  Xhat + xoff0 + d1);
    __builtin_nontemporal_store(Xr2,  Xhat + xoff0 + d2);
    __builtin_nontemporal_store(Xi1v, Xhat + xoff1 + d1);
    __builtin_nontemporal_store(Xi2v, Xhat + xoff1 + d2);
    A_lds[j * Dd + d1] = A1; A_lds[j * Dd + d2] = A2;
    sA1 += A1; sA2 += A2;
    Wr1 += A1 * Xr1;  Wr2 += A2 * Xr2;
    Wi1 += A1 * Xi1v; Wi2 += A2 * Xi2v;
  }

  psA[w * Dd + d1] = sA1; psA[w * Dd + d2] = sA2;
  pWr[w * Dd + d1] = Wr1; pWr[w * Dd + d2] = Wr2;
  pWi[w * Dd + d1] = Wi1; pWi[w * Dd + d2] = Wi2;
  __syncthreads();
  if (tid < Dd) {
    float sa = 0.f, wr = 0.f, wi = 0.f;
#pragma unroll
    for (int ww = 0; ww < 8; ++ww) {
      sa += psA[ww * Dd + tid];
      wr += pWr[ww * Dd + tid];
      wi += pWi[ww * Dd + tid];
    }
    const float inv = 1.0f / sa;
    invA[tid] = inv;
    EV[((size_t)(b * 2 + 0) * Ss + i) * Dd + tid] = wr * inv;   // est_v real
    EV[((size_t)(b * 2 + 1) * Ss + i) * Dd + tid] = wi * inv;   // est_v imag
  }
  __syncthreads();
  // pass 2: normalize A -> Qij. Lane-contiguous b64 LDS reads + b64 NT stores.
  const v2f iv = *(const v2f*)(invA + 2 * l);
  for (int j = w; j <= i; j += 8) {
    const v2f a = *(const v2f*)(A_lds + j * Dd + 2 * l);
    v2f r; r.x = a.x * iv.x; r.y = a.y * iv.y;
    __builtin_nontemporal_store(r, (v2f*)(Qij + (qijRow + j) * Dd + 2 * l));
  }
}

// ---------------------------------------------------------------------------
// est = W_e @ est_v + b_e ; pred = W_p @ rot(est_v) + b_p ; out = pred + est
// ---------------------------------------------------------------------------
__global__ void final_kernel(const float* __restrict__ EV,
                             const float* __restrict__ We, const float* __restrict__ be,
                             const float* __restrict__ Wp, const float* __restrict__ bp,
                             const float* __restrict__ tm, const float* __restrict__ lambda1,
                             float* __restrict__ est, float* __restrict__ outp) {
  const int lane = threadIdx.x & 31;
  const int wid  = (blockIdx.x * blockDim.x + threadIdx.x) >> 5;
  const int b       = wid >> 7;
  const int rowTile = (wid >> 2) & 31;
  const int colTile = wid & 3;
  const int half = lane >> 4;
  const int lr16 = lane & 15;
  const int m = rowTile * 16 + lr16;
  const int n = colTile * 16 + lr16;

  const float dtl = tm[b * (Ss + 1) + Ss] - tm[b * (Ss + 1) + Ss - 1];

  const float* EVr = EV + ((size_t)(b * 2 + 0) * Ss + m) * Dd;
  const float* EVi = EV + ((size_t)(b * 2 + 1) * Ss + m) * Dd;
  const float* Wer = We + (size_t)n * Dd;
  const float* Wei = We + (size_t)(Hh + n) * Dd;
  const float* Wpr = Wp + (size_t)n * Dd;
  const float* Wpi = Wp + (size_t)(Hh + n) * Dd;

  v8f eR = {}, eI = {}, pR = {}, pI = {};
  for (int k0 = 0; k0 < Dd; k0 += 4) {
    const int ka = k0 + half * 2;
    v2f ar = *(const v2f*)(EVr + ka);
    v2f ai = *(const v2f*)(EVi + ka);
    v2f mr, mi;
#pragma unroll
    for (int q = 0; q < 2; ++q) {          // per-K rotation coefficients
      const int kk = ka + q;
      const float rr = -fabsf(lambda1[kk & 31]);
      const float ii = (kk < 32) ? lambda1[32 + kk] : -lambda1[kk];
      const float e  = __expf(rr * dtl);
      float sn, cs; __sincosf(ii * dtl, &sn, &cs);
      mr[q] = e * cs; mi[q] = e * sn;
    }
    v2f pr = mr * ar - mi * ai;
    v2f pi = mr * ai + mi * ar;
    v2f ber = *(const v2f*)(Wer + ka);
    v2f bei = *(const v2f*)(Wei + ka);
    v2f bpr = *(const v2f*)(Wpr + ka);
    v2f bpi = *(const v2f*)(Wpi + ka);
    v2f nbei = -bei, nbpi = -bpi;
    eR = wmma4(ar, ber, eR); eR = wmma4(ai, nbei, eR);
    eI = wmma4(ar, bei, eI); eI = wmma4(ai, ber,  eI);
    pR = wmma4(pr, bpr, pR); pR = wmma4(pi, nbpi, pR);
    pI = wmma4(pr, bpi, pI); pI = wmma4(pi, bpr,  pI);
  }
  const float ber_ = be[n], bei_ = be[Hh + n];
  const float bpr_ = bp[n], bpi_ = bp[Hh + n];
#pragma unroll
  for (int v = 0; v < 8; ++v) {
    const int row = rowTile * 16 + (half ? v + 8 : v);
    const size_t o0 = ((size_t)(b * 2 + 0) * Ss + row) * Hh + n;
    const size_t o1 = ((size_t)(b * 2 + 1) * Ss + row) * Hh + n;
    const float er = eR[v] + ber_, ei = eI[v] + bei_;
    const float prv = pR[v] + bpr_, piv = pI[v] + bpi_;
    est[o0]  = er;  est[o1]  = ei;
    outp[o0] = DELTAc * prv + ETAc * er;
    outp[o1] = DELTAc * piv + ETAc * ei;
  }
}

__global__ void ri_kernel(const float* __restrict__ lambda1, float* __restrict__ outRI) {
  const int d = threadIdx.x;
  if (d < Dd) {
    outRI[d]      = -fabsf(lambda1[d & 31]);
    outRI[Dd + d] = (d < 32) ? lambda1[32 + d] : -lambda1[d];
  }
}

extern "C" void kernel_launch(void* const* d_in, const int* in_sizes, int n_in,
                              void* d_out, int out_size, void* d_ws, size_t ws_size,
                              hipStream_t stream) {
  const float* Xq = (const float*)d_in[0];
  const float* Xk = (const float*)d_in[1];
  const float* Xv = (const float*)d_in[2];
  const float* tm = (const float*)d_in[3];
  const float* lambda1 = (const float*)d_in[4];
  const float* lamOm  = (const float*)d_in[5];
  const float* lamOm0 = (const float*)d_in[6];
  const float* lamGam = (const float*)d_in[7];
  const float* lamC   = (const float*)d_in[8];
  const float* Wq = (const float*)d_in[9];
  const float* Wk = (const float*)d_in[10];
  const float* Wv = (const float*)d_in[11];
  const float* Wp = (const float*)d_in[12];
  const float* We = (const float*)d_in[13];
  const float* bq = (const float*)d_in[14];
  const float* bk = (const float*)d_in[15];
  const float* bv = (const float*)d_in[16];
  const float* bp = (const float*)d_in[17];
  const float* be = (const float*)d_in[18];

  float* out   = (float*)d_out;
  float* est   = out;                                   // B*2*S*H = 131072
  float* outp  = out + 131072;                          // 131072
  float* Qij   = out + 262144;                          // B*S*S*D = 33554432
  float* Xhat  = out + 262144 + 33554432;               // B*2*S*S*D = 67108864
  float* outRI = out + 262144 + 33554432 + 67108864;    // 128

  float* Qws  = (float*)d_ws;       // B*2*S*D each
  float* Kws  = Qws + 131072;
  float* Vws  = Kws + 131072;
  float* EVws = Vws + 131072;

  cproj_kernel<<<64, 128, 0, stream>>>(Xq, Wq, bq, Qws);
  cproj_kernel<<<64, 128, 0, stream>>>(Xk, Wk, bk, Kws);
  cproj_kernel<<<64, 128, 0, stream>>>(Xv, Wv, bv, Vws);

  const int smemBytes = (Ss * Dd + 3 * 8 * Dd + Dd) * (int)sizeof(float); // 137472 B (<320KB/WGP)
  hipFuncSetAttribute((const void*)attn_kernel,
                      hipFuncAttributeMaxDynamicSharedMemorySize, smemBytes);
  attn_kernel<<<Bb * Ss, 256, smemBytes, stream>>>(Qws, Kws, Vws, tm, lambda1,
                                                   lamOm, lamOm0, lamGam, lamC,
                                                   Qij, Xhat, EVws);

  final_kernel<<<64, 128, 0, stream>>>(EVws, We, be, Wp, bp, tm, lambda1, est, outp);
  ri_kernel<<<1, 64, 0, stream>>>(lambda1, outRI);
}